// BayesianTripletLoss_78082505441591
// MI455X (gfx1250) — compile-verified
//
#include <hip/hip_runtime.h>
#include <math.h>

typedef float v2f __attribute__((ext_vector_type(2)));
typedef float v8f __attribute__((ext_vector_type(8)));

#define BN 1024
#define DD 128

// ---------------------------------------------------------------------------
// Kernel 1: per-row squared norm of embeddings + per-row sum of clipped u
// ---------------------------------------------------------------------------
__global__ __launch_bounds__(128) void btl_rownorm(const float* __restrict__ E,
                                                   const float* __restrict__ U,
                                                   float* __restrict__ sq,
                                                   float* __restrict__ usum) {
  const int i = blockIdx.x;
  const int k = threadIdx.x;           // 0..127
  __shared__ float red0[4], red1[4];

  float e = E[(size_t)i * DD + k];
  float u = U[(size_t)i * DD + k];
  // clip to [1e-6, 1]; fmaxf(NaN,1e-6)=1e-6 matches the NaN->MIN_U sanitize
  u = fminf(fmaxf(u, 1e-6f), 1.0f);

  float s0 = e * e;
  float s1 = u;
  for (int off = 16; off >= 1; off >>= 1) {
    s0 += __shfl_xor(s0, off, 32);
    s1 += __shfl_xor(s1, off, 32);
  }
  const int wave = k >> 5, lane = k & 31;
  if (lane == 0) { red0[wave] = s0; red1[wave] = s1; }
  __syncthreads();
  if (k == 0) {
    sq[i]   = red0[0] + red0[1] + red0[2] + red0[3];
    usum[i] = red1[0] + red1[1] + red1[2] + red1[3];
  }
}

// ---------------------------------------------------------------------------
// Kernel 2: Gram matrix via V_WMMA_F32_16X16X4_F32, fused with hardest-pos /
// hardest-neg mining. One block per 16-row tile; 8 waves sweep 64 col tiles.
// ---------------------------------------------------------------------------
__global__ __launch_bounds__(256) void btl_gram_mine(const float* __restrict__ E,
                                                     const int*   __restrict__ labels,
                                                     const float* __restrict__ sq,
                                                     float* __restrict__ dpos,
                                                     float* __restrict__ dneg,
                                                     int*   __restrict__ posidx,
                                                     int*   __restrict__ negidx,
                                                     int*   __restrict__ validf) {
  __shared__ int   s_lab[BN];
  __shared__ float s_sq[BN];
  __shared__ float s_pv[8][16];
  __shared__ int   s_pi[8][16];
  __shared__ float s_nv[8][16];
  __shared__ int   s_ni[8][16];

  const int tid  = threadIdx.x;
  const int wave = tid >> 5;
  const int lane = tid & 31;
  const int half = lane >> 4;          // 0: lanes 0-15, 1: lanes 16-31
  const int l16  = lane & 15;
  const int rowBase = blockIdx.x * 16;

  for (int i = tid; i < BN; i += 256) { s_lab[i] = labels[i]; s_sq[i] = sq[i]; }
  __syncthreads();

  // per-lane mining state for the 8 C-rows this lane owns
  float bestPv[8], bestNv[8];
  int   bestPi[8], bestNi[8];
#pragma unroll
  for (int r = 0; r < 8; ++r) {
    bestPv[r] = -1e30f; bestPi[r] = 0x7fffffff;
    bestNv[r] =  1e30f; bestNi[r] = 0x7fffffff;
  }

  // A fragment source: row rowBase+l16, K pair selected by lane half (ISA 7.12.2)
  const float* arow = E + (size_t)(rowBase + l16) * DD + 2 * half;

  for (int ct = wave; ct < BN / 16; ct += 8) {
    const int colBase = ct * 16;
    const float* brow = E + (size_t)(colBase + l16) * DD + 2 * half;

    v8f c = {0.f, 0.f, 0.f, 0.f, 0.f, 0.f, 0.f, 0.f};
#pragma unroll
    for (int k = 0; k < DD; k += 4) {
      v2f a, b;
      a.x = arow[k]; a.y = arow[k + 1];
      b.x = brow[k]; b.y = brow[k + 1];
      // D = A(16x4,f32) * B(4x16,f32) + C ; full fp32 matrix op
      c = __builtin_amdgcn_wmma_f32_16x16x4_f32(false, a, false, b,
                                                (short)0, c, false, false);
    }

    const int   n    = colBase + l16;    // this lane's column for all c[r]
    const int   labn = s_lab[n];
    const float sqn  = s_sq[n];
#pragma unroll
    for (int r = 0; r < 8; ++r) {
      const int mrow = rowBase + r + 8 * half;
      float d2 = s_sq[mrow] + sqn - 2.0f * c[r];
      float d  = sqrtf(fmaxf(d2, 0.0f)) + 1e-8f;
      bool same = (s_lab[mrow] == labn);
      if (same) {
        if (mrow != n) {
          if (d > bestPv[r] || (d == bestPv[r] && n < bestPi[r])) {
            bestPv[r] = d; bestPi[r] = n;
          }
        }
      } else {
        if (d < bestNv[r] || (d == bestNv[r] && n < bestNi[r])) {
          bestNv[r] = d; bestNi[r] = n;
        }
      }
    }
  }

  // reduce across the 16 lanes of each half (xor offsets 1..8 stay in-half)
#pragma unroll
  for (int r = 0; r < 8; ++r) {
    for (int off = 8; off >= 1; off >>= 1) {
      float ov = __shfl_xor(bestPv[r], off, 32);
      int   oi = __shfl_xor(bestPi[r], off, 32);
      if (ov > bestPv[r] || (ov == bestPv[r] && oi < bestPi[r])) { bestPv[r] = ov; bestPi[r] = oi; }
      ov = __shfl_xor(bestNv[r], off, 32);
      oi = __shfl_xor(bestNi[r], off, 32);
      if (ov < bestNv[r] || (ov == bestNv[r] && oi < bestNi[r])) { bestNv[r] = ov; bestNi[r] = oi; }
    }
  }
  if (l16 == 0) {                       // lanes 0 and 16 publish 8 rows each
#pragma unroll
    for (int r = 0; r < 8; ++r) {
      s_pv[wave][r + 8 * half] = bestPv[r];
      s_pi[wave][r + 8 * half] = bestPi[r];
      s_nv[wave][r + 8 * half] = bestNv[r];
      s_ni[wave][r + 8 * half] = bestNi[r];
    }
  }
  __syncthreads();

  if (tid < 16) {                       // combine the 8 waves, one row per thread
    float pv = -1e30f; int pi = 0x7fffffff;
    float nv =  1e30f; int ni = 0x7fffffff;
    for (int w = 0; w < 8; ++w) {
      float v = s_pv[w][tid]; int ix = s_pi[w][tid];
      if (v > pv || (v == pv && ix < pi)) { pv = v; pi = ix; }
      v = s_nv[w][tid]; ix = s_ni[w][tid];
      if (v < nv || (v == nv && ix < ni)) { nv = v; ni = ix; }
    }
    const int row = rowBase + tid;
    const bool fP = (pv > -1e29f);
    const bool fN = (nv <  1e29f);
    dpos[row]   = pv;
    dneg[row]   = nv;
    posidx[row] = fP ? pi : 0;
    negidx[row] = fN ? ni : 0;
    validf[row] = (fP && fN) ? 1 : 0;
  }
}

// ---------------------------------------------------------------------------
// Kernel 3: per-anchor du at the two mined pairs + per-triplet loss
// ---------------------------------------------------------------------------
__global__ __launch_bounds__(128) void btl_triplet(const float* __restrict__ E,
                                                   const float* __restrict__ U,
                                                   const float* __restrict__ dpos,
                                                   const float* __restrict__ dneg,
                                                   const int*   __restrict__ posidx,
                                                   const int*   __restrict__ negidx,
                                                   const int*   __restrict__ validf,
                                                   float* __restrict__ lossArr) {
  const int i = blockIdx.x;
  const int k = threadIdx.x;            // 0..127
  __shared__ float redp[4], redn[4];

  if (!validf[i]) {                     // uniform per block
    if (k == 0) lossArr[i] = 0.0f;
    return;
  }
  const int p = posidx[i], q = negidx[i];
  const float dp = dpos[i], dn = dneg[i];

  float ei = E[(size_t)i * DD + k];
  float u  = U[(size_t)i * DD + k];
  u = fminf(fmaxf(u, 1e-6f), 1.0f);
  float u2 = u * u;

  float fp = (ei - E[(size_t)p * DD + k]) / dp;
  float fn = (ei - E[(size_t)q * DD + k]) / dn;
  float sp = fp * fp * u2;
  float sn = fn * fn * u2;
  for (int off = 16; off >= 1; off >>= 1) {
    sp += __shfl_xor(sp, off, 32);
    sn += __shfl_xor(sn, off, 32);
  }
  const int wave = k >> 5, lane = k & 31;
  if (lane == 0) { redp[wave] = sp; redn[wave] = sn; }
  __syncthreads();
  if (k == 0) {
    sp = redp[0] + redp[1] + redp[2] + redp[3];
    sn = redn[0] + redn[1] + redn[2] + redn[3];
    float up = sqrtf(sp + 1e-8f);
    float un = sqrtf(sn + 1e-8f);
    float sigma  = sqrtf(up * up + un * un + 1e-8f);
    float margin = 0.3f * (1.0f + sigma);          // adaptive margin
    float raw    = dp - dn + margin;               // TEMPERATURE = 1
    float x      = 10.0f * raw;                    // LOSS_SCALE
    float spl    = fmaxf(x, 0.0f) + log1pf(expf(-fabsf(x)));  // softplus
    lossArr[i] = spl * 0.1f;
  }
}

// ---------------------------------------------------------------------------
// Kernel 4: final reduction -> scalar loss
// ---------------------------------------------------------------------------
__global__ __launch_bounds__(256) void btl_finalize(const float* __restrict__ lossArr,
                                                    const int*   __restrict__ validf,
                                                    const float* __restrict__ usum,
                                                    float* __restrict__ out) {
  __shared__ float rl[8], rv[8], ru[8];
  float ls = 0.f, vc = 0.f, us = 0.f;
  for (int i = threadIdx.x; i < BN; i += 256) {
    ls += lossArr[i];
    vc += (float)validf[i];
    us += usum[i];
  }
  for (int off = 16; off >= 1; off >>= 1) {
    ls += __shfl_xor(ls, off, 32);
    vc += __shfl_xor(vc, off, 32);
    us += __shfl_xor(us, off, 32);
  }
  const int wave = threadIdx.x >> 5, lane = threadIdx.x & 31;
  if (lane == 0) { rl[wave] = ls; rv[wave] = vc; ru[wave] = us; }
  __syncthreads();
  if (threadIdx.x == 0) {
    ls = 0.f; vc = 0.f; us = 0.f;
    for (int w = 0; w < 8; ++w) { ls += rl[w]; vc += rv[w]; us += ru[w]; }
    float nvalid = fmaxf(vc, 1.0f);
    float reg = us / ((float)BN * (float)DD);
    out[0] = ls / nvalid + 0.05f * reg;
  }
}

// ---------------------------------------------------------------------------
extern "C" void kernel_launch(void* const* d_in, const int* in_sizes, int n_in,
                              void* d_out, int out_size, void* d_ws, size_t ws_size,
                              hipStream_t stream) {
  const float* E      = (const float*)d_in[0];   // embeddings [1024,128] f32
  const float* U      = (const float*)d_in[1];   // uncertainties [1024,128] f32
  const int*   labels = (const int*)d_in[2];     // labels [1024] i32
  float* out = (float*)d_out;

  float* w      = (float*)d_ws;
  float* sq     = w;                 // 1024 f
  float* usum   = w + 1024;          // 1024 f
  float* dpos   = w + 2048;          // 1024 f
  float* dneg   = w + 3072;          // 1024 f
  int*   posidx = (int*)(w + 4096);  // 1024 i
  int*   negidx = (int*)(w + 5120);  // 1024 i
  int*   validf = (int*)(w + 6144);  // 1024 i
  float* lossA  = w + 7168;          // 1024 f

  btl_rownorm  <<<BN,       128, 0, stream>>>(E, U, sq, usum);
  btl_gram_mine<<<BN / 16,  256, 0, stream>>>(E, labels, sq, dpos, dneg,
                                              posidx, negidx, validf);
  btl_triplet  <<<BN,       128, 0, stream>>>(E, U, dpos, dneg, posidx, negidx,
                                              validf, lossA);
  btl_finalize <<<1,        256, 0, stream>>>(lossA, validf, usum, out);
}